// Model_47656957116899
// MI455X (gfx1250) — compile-verified
//
#include <hip/hip_runtime.h>
#include <hip/hip_bf16.h>

typedef __attribute__((ext_vector_type(2))) float v2f;
typedef __attribute__((ext_vector_type(8))) float v8f;

#define GNN_N 100000
#define GNN_E 1600000
#define GNN_D 128

// ---------------------------------------------------------------------------
// Zero a float buffer (float4 grid-stride; count must be a multiple of 4).
// ---------------------------------------------------------------------------
__global__ void zero_f4(float* __restrict__ p, size_t n_floats) {
    size_t n4 = n_floats >> 2;
    float4 z = make_float4(0.f, 0.f, 0.f, 0.f);
    for (size_t i = (size_t)blockIdx.x * blockDim.x + threadIdx.x; i < n4;
         i += (size_t)gridDim.x * blockDim.x) {
        ((float4*)p)[i] = z;
    }
}

// ---------------------------------------------------------------------------
// Degree count: cnt[dst[e]] += 1.0f
// ---------------------------------------------------------------------------
__global__ void count_deg(const int* __restrict__ dst, float* __restrict__ cnt,
                          int n_edges) {
    int e = blockIdx.x * blockDim.x + threadIdx.x;
    if (e < n_edges) atomicAdd(&cnt[dst[e]], 1.0f);
}

// ---------------------------------------------------------------------------
// invdeg[i] = 1 / max(cnt[i], 1)
// ---------------------------------------------------------------------------
__global__ void inv_deg(const float* __restrict__ cnt, float* __restrict__ invdeg,
                        int n) {
    int i = blockIdx.x * blockDim.x + threadIdx.x;
    if (i < n) invdeg[i] = 1.0f / fmaxf(cnt[i], 1.0f);
}

// ---------------------------------------------------------------------------
// Scatter-add: one wave32 per edge; lane handles 4 consecutive floats.
// feat rows are L2-resident (51 MB) so the gather hits L2; atomics land in L2.
// ---------------------------------------------------------------------------
__global__ void scatter_add(const float* __restrict__ feat,
                            const int* __restrict__ src,
                            const int* __restrict__ dst,
                            float* __restrict__ agg, int n_edges) {
    int e = (int)(((size_t)blockIdx.x * blockDim.x + threadIdx.x) >> 5);
    if (e >= n_edges) return;
    int lane = threadIdx.x & 31;
    int s = src[e];
    int d = dst[e];
    float4 v = ((const float4*)(feat + (size_t)s * GNN_D))[lane];
    float* o = agg + (size_t)d * GNN_D + lane * 4;
    atomicAdd(o + 0, v.x);
    atomicAdd(o + 1, v.y);
    atomicAdd(o + 2, v.z);
    atomicAdd(o + 3, v.w);
}

// ---------------------------------------------------------------------------
// Fused SAGEConv GEMM:  out = act( (agg .* invdeg) @ Wl + b + xroot @ Wr )
//
// Block = 256 threads = 8 wave32. Weights (128x128 each), bias, and the
// 16-row A tiles live in dynamic LDS (~145 KB; 320 KB/WGP on CDNA5).
// Each wave computes one 16x16 output tile via V_WMMA_F32_16X16X4_F32.
// Two independent accumulator chains (Wl-path / Wr-path) so the 64 WMMAs
// per tile form two interleavable 32-deep D->C chains instead of one
// 64-deep serial chain, hiding XDL accumulate latency.
// ---------------------------------------------------------------------------
__global__ __launch_bounds__(256, 1)
void sage_gemm_wmma(const float* __restrict__ agg,
                    const float* __restrict__ invdeg,
                    const float* __restrict__ xroot,
                    const float* __restrict__ Wl,
                    const float* __restrict__ Wr,
                    const float* __restrict__ bias,
                    float* __restrict__ out, int n_nodes, int relu_flag) {
    extern __shared__ float smem[];
    float* sWl = smem;              // 128*128 = 16384 floats
    float* sWr = sWl + GNN_D * GNN_D;
    float* sA  = sWr + GNN_D * GNN_D;  // 16*128 = 2048 floats (mean tile)
    float* sX  = sA + 16 * GNN_D;      // 16*128 (root tile)
    float* sB  = sX + 16 * GNN_D;      // 128 (bias)

    // Stage weights + bias once per block (L2-resident source).
    for (int i = threadIdx.x; i < (GNN_D * GNN_D) / 4; i += 256) {
        ((float4*)sWl)[i] = ((const float4*)Wl)[i];
        ((float4*)sWr)[i] = ((const float4*)Wr)[i];
    }
    if (threadIdx.x < GNN_D) sB[threadIdx.x] = bias[threadIdx.x];

    const int lane  = threadIdx.x & 31;
    const int wave  = threadIdx.x >> 5;  // 0..7 -> 16-col slice
    const int m     = lane & 15;         // A row / B-D col within tile
    const int khalf = (lane >> 4) << 1;  // 0 or 2 (A/B K split across half-waves)
    const int n0    = wave * 16;

    const int numTiles = n_nodes >> 4;  // 100000/16 = 6250
    for (int t = blockIdx.x; t < numTiles; t += gridDim.x) {
        const int r0 = t << 4;

        __syncthreads();  // weights ready / previous tile's sA,sX consumed
        // Stage 16-row A tiles; fold invdeg into the mean tile.
        for (int i = threadIdx.x; i < 16 * (GNN_D / 4); i += 256) {
            int row = i >> 5;   // 32 float4 per row
            int c4  = i & 31;
            float inv = invdeg[r0 + row];
            float4 a = ((const float4*)(agg + (size_t)(r0 + row) * GNN_D))[c4];
            a.x *= inv; a.y *= inv; a.z *= inv; a.w *= inv;
            ((float4*)sA)[i] = a;
            ((float4*)sX)[i] = ((const float4*)(xroot + (size_t)(r0 + row) * GNN_D))[c4];
        }
        __syncthreads();

        v8f accL = {};  // (agg.*invdeg) @ Wl chain
        v8f accR = {};  // xroot @ Wr chain (independent of accL)
#pragma unroll
        for (int k0 = 0; k0 < GNN_D; k0 += 4) {
            const int k = k0 + khalf;
            // A fragments: 16x4 f32, row m, K = k, k+1 (ds_load_b64)
            v2f aL = *(const v2f*)(sA + m * GNN_D + k);
            v2f aR = *(const v2f*)(sX + m * GNN_D + k);
            // B fragments: 4x16 f32, K rows striped across lanes, col n0+m
            v2f bL; bL.x = sWl[k * GNN_D + n0 + m]; bL.y = sWl[(k + 1) * GNN_D + n0 + m];
            v2f bR; bR.x = sWr[k * GNN_D + n0 + m]; bR.y = sWr[(k + 1) * GNN_D + n0 + m];
            accL = __builtin_amdgcn_wmma_f32_16x16x4_f32(
                false, aL, false, bL, (short)0, accL, false, false);
            accR = __builtin_amdgcn_wmma_f32_16x16x4_f32(
                false, aR, false, bR, (short)0, accR, false, false);
        }

        // Epilogue: merge chains + bias + optional ReLU, store per C/D layout
        // (VGPR i: lanes 0-15 -> row i, lanes 16-31 -> row i+8).
        const float bn = sB[n0 + m];
        const int rbase = r0 + ((lane >> 4) << 3);
#pragma unroll
        for (int i = 0; i < 8; i++) {
            float v = accL[i] + accR[i] + bn;
            if (relu_flag) v = fmaxf(v, 0.0f);
            out[(size_t)(rbase + i) * GNN_D + n0 + m] = v;
        }
    }
}

// ---------------------------------------------------------------------------
extern "C" void kernel_launch(void* const* d_in, const int* in_sizes, int n_in,
                              void* d_out, int out_size, void* d_ws, size_t ws_size,
                              hipStream_t stream) {
    const float* x   = (const float*)d_in[0];
    const int*   ei  = (const int*)d_in[1];   // [2, E] flat: src then dst
    const float* W1l = (const float*)d_in[2];
    const float* b1  = (const float*)d_in[3];
    const float* W1r = (const float*)d_in[4];
    const float* W2l = (const float*)d_in[5];
    const float* b2  = (const float*)d_in[6];
    const float* W2r = (const float*)d_in[7];
    float* out = (float*)d_out;

    const int N = GNN_N, E = GNN_E, D = GNN_D;
    const int* src = ei;
    const int* dst = ei + E;

    // Workspace layout: [agg N*D][cnt N][invdeg N][h N*D]  (~103 MB)
    float* agg    = (float*)d_ws;
    float* cnt    = agg + (size_t)N * D;
    float* invdeg = cnt + N;
    float* h      = invdeg + N;

    const size_t shmem = (size_t)(2 * D * D + 2 * 16 * D + D) * sizeof(float);

    // ---- Layer 1 ----
    zero_f4<<<4096, 256, 0, stream>>>(agg, (size_t)N * D + N);  // agg + cnt
    count_deg<<<(E + 255) / 256, 256, 0, stream>>>(dst, cnt, E);
    inv_deg<<<(N + 255) / 256, 256, 0, stream>>>(cnt, invdeg, N);
    scatter_add<<<E / 8, 256, 0, stream>>>(x, src, dst, agg, E);
    sage_gemm_wmma<<<2048, 256, shmem, stream>>>(agg, invdeg, x, W1l, W1r, b1,
                                                 h, N, /*relu=*/1);

    // ---- Layer 2 ----
    zero_f4<<<4096, 256, 0, stream>>>(agg, (size_t)N * D);
    scatter_add<<<E / 8, 256, 0, stream>>>(h, src, dst, agg, E);
    sage_gemm_wmma<<<2048, 256, shmem, stream>>>(agg, invdeg, h, W2l, W2r, b2,
                                                 out, N, /*relu=*/0);
}